// BOW_model_41291815583753
// MI455X (gfx1250) — compile-verified
//
#include <hip/hip_runtime.h>
#include <hip/hip_bf16.h>
#include <math.h>

#define B_DIM 4096
#define L_DIM 200
#define V_DIM 100000
#define D_DIM 256
#define BN_EPS 1e-5f

typedef float v2f __attribute__((ext_vector_type(2)));
typedef float v8f __attribute__((ext_vector_type(8)));

// ---------------------------------------------------------------------------
// Kernel 1: EmbeddingBag-mean pooling.
// block = one batch row (256 threads, thread = feature d). Tokens staged in
// LDS; emb rows gathered coalesced (1KB/row); prefetch 4 rows ahead so the
// L2 (table fits: 100MB < 192MB) stream stays full.
// ---------------------------------------------------------------------------
__global__ void embed_pool_kernel(const int* __restrict__ tokens,
                                  const int* __restrict__ lengths,
                                  const float* __restrict__ emb,
                                  float* __restrict__ pooled) {
    const int b = blockIdx.x;
    const int d = threadIdx.x;
    __shared__ int toks[L_DIM];
    if (d < L_DIM) toks[d] = tokens[(size_t)b * L_DIM + d];
    __syncthreads();

    const int len = lengths[b];
    float acc = 0.0f;
    for (int l = 0; l < len; ++l) {
        if (l + 4 < len) {
            const float* pf = emb + (size_t)toks[l + 4] * D_DIM + d;
            __builtin_prefetch(pf, 0, 3);   // -> global_prefetch_b8
        }
        acc += emb[(size_t)toks[l] * D_DIM + d];
    }
    pooled[(size_t)b * D_DIM + d] = acc / (float)len;
}

// ---------------------------------------------------------------------------
// Kernel 2: z = pooled @ W1^T + b1 via V_WMMA_F32_16X16X4_F32 (fp32 exact).
// Block = 512 threads = 16 waves; wave w computes the 16x16 tile at
// (m0 = blockIdx.x*16, n0 = w*16). K=256 stepped by 4.
// A frag (16x4): lane&15 = M row, lane>>4 selects K-pair -> contiguous float2.
// B frag (4x16): B[k,n] = W1[n,k]; lane&15 = N col, lane>>4 = K-pair -> float2.
// C/D (16x16): VGPR r holds row M = r + 8*(lane>>4), col N = lane&15.
// ---------------------------------------------------------------------------
__global__ void gemm_wmma_kernel(const float* __restrict__ pooled,
                                 const float* __restrict__ W1,
                                 const float* __restrict__ b1,
                                 float* __restrict__ z) {
    const int lane = threadIdx.x & 31;
    const int wave = threadIdx.x >> 5;
    const int col  = lane & 15;
    const int half = lane >> 4;
    const int m0 = blockIdx.x * 16;
    const int n0 = wave * 16;

    const float* Ap = pooled + (size_t)(m0 + col) * D_DIM + 2 * half;
    const float* Bp = W1     + (size_t)(n0 + col) * D_DIM + 2 * half;

    v8f acc0 = {};
    v8f acc1 = {};
    #pragma unroll 8
    for (int k0 = 0; k0 < D_DIM; k0 += 8) {
        v2f a0 = *(const v2f*)(Ap + k0);
        v2f b0 = *(const v2f*)(Bp + k0);
        acc0 = __builtin_amdgcn_wmma_f32_16x16x4_f32(false, a0, false, b0,
                                                     (short)0, acc0, false, false);
        v2f a1 = *(const v2f*)(Ap + k0 + 4);
        v2f b1v = *(const v2f*)(Bp + k0 + 4);
        acc1 = __builtin_amdgcn_wmma_f32_16x16x4_f32(false, a1, false, b1v,
                                                     (short)0, acc1, false, false);
    }

    const float bias = b1[n0 + col];
    #pragma unroll
    for (int r = 0; r < 8; ++r) {
        const int m = m0 + r + 8 * half;
        z[(size_t)m * D_DIM + n0 + col] = acc0[r] + acc1[r] + bias;
    }
}

// ---------------------------------------------------------------------------
// Kernel 3a: BN partial sums (coalesced: thread = feature, rows chunked per
// block), fp32 atomics into 256-entry sum/sumsq.
// ---------------------------------------------------------------------------
#define BN_ROWS 64
__global__ void bn_partial_kernel(const float* __restrict__ z,
                                  float* __restrict__ sum,
                                  float* __restrict__ sumsq) {
    const int d = threadIdx.x;
    const int r0 = blockIdx.x * BN_ROWS;
    float s = 0.0f, s2 = 0.0f;
    #pragma unroll 4
    for (int r = 0; r < BN_ROWS; ++r) {
        float v = z[(size_t)(r0 + r) * D_DIM + d];
        s  += v;
        s2 += v * v;
    }
    atomicAdd(&sum[d], s);
    atomicAdd(&sumsq[d], s2);
}

// Kernel 3b: finalize mu / rstd (one block, 256 threads).
__global__ void bn_finalize_kernel(const float* __restrict__ sum,
                                   const float* __restrict__ sumsq,
                                   float* __restrict__ mu,
                                   float* __restrict__ rstd) {
    const int d = threadIdx.x;
    const float m = sum[d] * (1.0f / B_DIM);
    const float var = sumsq[d] * (1.0f / B_DIM) - m * m;
    mu[d] = m;
    rstd[d] = rsqrtf(var + BN_EPS);
}

// ---------------------------------------------------------------------------
// Kernel 4: normalize + ReLU + dot(w2) -> logits[b]; BCE term atomically
// accumulated into d_out[0] (pre-zeroed by hipMemsetAsync).
// ---------------------------------------------------------------------------
__global__ void head_kernel(const float* __restrict__ z,
                            const float* __restrict__ mu,
                            const float* __restrict__ rstd,
                            const float* __restrict__ gamma,
                            const float* __restrict__ beta,
                            const float* __restrict__ w2,
                            const float* __restrict__ b2,
                            const float* __restrict__ t,
                            float* __restrict__ out) {
    const int b = blockIdx.x;
    const int d = threadIdx.x;
    __shared__ float red[D_DIM];

    float v = z[(size_t)b * D_DIM + d];
    float h = gamma[d] * (v - mu[d]) * rstd[d] + beta[d];
    h = fmaxf(h, 0.0f);
    red[d] = h * w2[d];
    __syncthreads();
    for (int s = D_DIM / 2; s > 0; s >>= 1) {
        if (d < s) red[d] += red[d + s];
        __syncthreads();
    }
    if (d == 0) {
        const float logit = red[0] + b2[0];
        out[1 + b] = logit;
        const float term = fmaxf(logit, 0.0f) - logit * t[b]
                         + log1pf(expf(-fabsf(logit)));
        atomicAdd(out, term * (1.0f / B_DIM));
    }
}

// ---------------------------------------------------------------------------
extern "C" void kernel_launch(void* const* d_in, const int* in_sizes, int n_in,
                              void* d_out, int out_size, void* d_ws, size_t ws_size,
                              hipStream_t stream) {
    const int*   tokens  = (const int*)d_in[0];
    const int*   lengths = (const int*)d_in[1];
    const float* t       = (const float*)d_in[2];
    const float* emb     = (const float*)d_in[3];
    const float* W1      = (const float*)d_in[4];
    const float* b1      = (const float*)d_in[5];
    const float* gamma   = (const float*)d_in[6];
    const float* beta    = (const float*)d_in[7];
    const float* w2      = (const float*)d_in[8];
    const float* b2      = (const float*)d_in[9];
    float* out = (float*)d_out;

    float* ws     = (float*)d_ws;
    float* pooled = ws;                                   // 4096*256
    float* z      = pooled + (size_t)B_DIM * D_DIM;       // 4096*256
    float* sum    = z + (size_t)B_DIM * D_DIM;            // 256
    float* sumsq  = sum + D_DIM;                          // 256
    float* mu     = sumsq + D_DIM;                        // 256
    float* rstd   = mu + D_DIM;                           // 256

    // zero the atomic accumulators (graph-capture-safe)
    hipMemsetAsync(sum, 0, 2 * D_DIM * sizeof(float), stream);
    hipMemsetAsync(out, 0, sizeof(float), stream);

    embed_pool_kernel<<<B_DIM, D_DIM, 0, stream>>>(tokens, lengths, emb, pooled);
    gemm_wmma_kernel<<<B_DIM / 16, 512, 0, stream>>>(pooled, W1, b1, z);
    bn_partial_kernel<<<B_DIM / BN_ROWS, D_DIM, 0, stream>>>(z, sum, sumsq);
    bn_finalize_kernel<<<1, D_DIM, 0, stream>>>(sum, sumsq, mu, rstd);
    head_kernel<<<B_DIM, D_DIM, 0, stream>>>(z, mu, rstd, gamma, beta, w2, b2, t, out);
}